// GALExtrapolation_14654428414227
// MI455X (gfx1250) — compile-verified
//
#include <hip/hip_runtime.h>

#define N_NODES 20000
#define T_IN    12
#define N_IN    16
#define KP1     9

typedef float v2f __attribute__((ext_vector_type(2)));
typedef float v8f __attribute__((ext_vector_type(8)));

// ---------------------------------------------------------------------------
// Fold heads + EMA-invariant parts into a 12x12 matrix W2 and bias b2 (exact):
//   W2[tt][t] = sum_h mh_w[h] * shrink_w[tt][t*8+h]
//   b2[tt]    = shrink_b[tt] + sum_{t,h} mh_b[h] * shrink_w[tt][t*8+h]
// ws layout: ws[0..143] = W2 row-major [tt][t], ws[144..155] = b2
// ---------------------------------------------------------------------------
__global__ __launch_bounds__(256) void prep_w2(const float* __restrict__ shrink_w,
                                               const float* __restrict__ shrink_b,
                                               const float* __restrict__ mh_w,
                                               const float* __restrict__ mh_b,
                                               float* __restrict__ ws) {
  const int tid = threadIdx.x;
  if (tid < 144) {
    const int tt = tid / 12, t = tid % 12;
    float acc = 0.f;
    for (int h = 0; h < 8; ++h) acc = fmaf(mh_w[h], shrink_w[tt * 96 + t * 8 + h], acc);
    ws[tt * 12 + t] = acc;
  } else if (tid < 156) {
    const int tt = tid - 144;
    float acc = shrink_b[tt];
    for (int t = 0; t < 12; ++t)
      for (int h = 0; h < 8; ++h) acc = fmaf(mh_b[h], shrink_w[tt * 96 + t * 8 + h], acc);
    ws[144 + tt] = acc;
  }
}

// ---------------------------------------------------------------------------
// out[:, :12] = x   (float4 copy; 15000*256 threads == 3,840,000 float4 exactly)
// ---------------------------------------------------------------------------
__global__ __launch_bounds__(256) void copy_head(const float4* __restrict__ x4,
                                                 float4* __restrict__ out4) {
  const int i   = blockIdx.x * 256 + threadIdx.x;   // [0, 3,840,000)
  const int bt  = i / 80000;                        // plane = 20000*16/4 float4
  const int off = i - bt * 80000;
  const int b   = bt / 12, t = bt - b * 12;
  out4[(size_t)(b * 24 + t) * 80000 + off] = x4[i];
}

// ---------------------------------------------------------------------------
// Main fused kernel: one wave per (b, node). 80,000 waves total (grid exact,
// EXEC all-ones for WMMA). Gather->swish->EMA, then 16x16 tile GEMM (K=12)
// via 3x V_WMMA_F32_16X16X4_F32, then selu + coalesced b128 stores.
// ---------------------------------------------------------------------------
__global__ __launch_bounds__(256) void gal_main(const float* __restrict__ x,
                                                const int* __restrict__ nn,
                                                const float* __restrict__ agg_w,
                                                const float* __restrict__ agg_b,
                                                const float* __restrict__ w2b2,
                                                float* __restrict__ out) {
  const int wave = (int)((blockIdx.x * 256u + threadIdx.x) >> 5);  // 0..79999
  const int lane = threadIdx.x & 31;
  const int half = lane >> 4;    // WMMA lane-half (K split)
  const int c    = lane & 15;    // A: channel row / B,C,D: column index
  const int b    = wave / N_NODES;
  const int node = wave - b * N_NODES;

  // neighbor indices (uniform per wave) + aggregation weights
  int idx[KP1];
#pragma unroll
  for (int k = 0; k < KP1; ++k) idx[k] = nn[node * KP1 + k];
  float aw[KP1];
#pragma unroll
  for (int k = 0; k < KP1; ++k) aw[k] = agg_w[k];
  const float ab = agg_b[0];

  // s[j] = swish(gather-agg) at t_j = 4*(j>>1) + 2*half + (j&1)
  // (the exact t slots this lane must supply to the 3 WMMA A-operands)
  float s[6];
#pragma unroll
  for (int j = 0; j < 6; ++j) {
    const int t = 4 * (j >> 1) + 2 * half + (j & 1);
    const float* __restrict__ xp = x + (size_t)(b * T_IN + t) * (N_NODES * N_IN) + c;
    float acc = ab;
#pragma unroll
    for (int k = 0; k < KP1; ++k) {
      const float v = (idx[k] < N_NODES) ? xp[(size_t)idx[k] * N_IN] : 0.f;  // pad row = 0
      acc = fmaf(aw[k], v, acc);
    }
    // swish (beta=0.8): acc * sigmoid(0.8*acc); v_rcp_f32 instead of IEEE divide
    s[j] = acc * __builtin_amdgcn_rcpf(1.f + __expf(-0.8f * acc));
  }

  // EMA: u[t] = t==0 ? s[0] : 0.8 s[t] + 0.2 s[t-1]; s[t-1] via lane^16 shuffle
  float other[6];
#pragma unroll
  for (int j = 0; j < 6; ++j) other[j] = __shfl_xor(s[j], 16, 32);
  float u[6];
#pragma unroll
  for (int j = 0; j < 6; ++j) {
    float prev;
    if (j & 1)          prev = s[j - 1];                       // own previous slot
    else if (half == 0) prev = (j == 0) ? 0.f : other[j - 1];  // from half1
    else                prev = other[j + 1];                   // from half0
    float val = fmaf(0.2f, prev, 0.8f * s[j]);
    if (half == 0 && j == 0) val = s[0];                       // t = 0: no EMA
    u[j] = val;
  }

  // B operand: W2^T with same per-lane t slots; pad columns 12..15 with 0
  float w[6];
#pragma unroll
  for (int j = 0; j < 6; ++j) {
    const int t = 4 * (j >> 1) + 2 * half + (j & 1);
    w[j] = (c < 12) ? w2b2[c * 12 + t] : 0.f;
  }

  // D(16x16) = A(16x12) x B(12x16) via 3x v_wmma_f32_16x16x4_f32
  v8f acc = {0.f, 0.f, 0.f, 0.f, 0.f, 0.f, 0.f, 0.f};
#pragma unroll
  for (int i = 0; i < 3; ++i) {
    v2f av; av.x = u[2 * i]; av.y = u[2 * i + 1];
    v2f bv; bv.x = w[2 * i]; bv.y = w[2 * i + 1];
#if defined(__HIP_DEVICE_COMPILE__) && __has_builtin(__builtin_amdgcn_wmma_f32_16x16x4_f32)
    acc = __builtin_amdgcn_wmma_f32_16x16x4_f32(false, av, false, bv, (short)0, acc,
                                                false, false);
#else
    acc[0] = fmaf(av.x, bv.x, acc[0]);  // host pass placeholder only
#endif
  }

  // Epilogue: lane holds y[ch = r + 8*half][tt = c] in acc[r]; selu + store.
  // Branchless selu: exp always evaluated (clamped arg), then value-select ->
  // straight-line v_exp_f32 + v_cndmask instead of exec-mask branches.
  if (c < 12) {
    const float bias = w2b2[144 + c];
    float* __restrict__ op =
        out + ((size_t)(b * 24 + T_IN + c) * N_NODES + node) * N_IN + 8 * half;
    float vals[8];
#pragma unroll
    for (int r = 0; r < 8; ++r) {
      const float v   = acc[r] + bias;
      const float neg = 1.7580993408473766f * (__expf(fminf(v, 0.f)) - 1.f);
      const float pos = 1.0507009873554805f * v;
      vals[r] = (v > 0.f) ? pos : neg;
    }
    *(float4*)(op)     = make_float4(vals[0], vals[1], vals[2], vals[3]);
    *(float4*)(op + 4) = make_float4(vals[4], vals[5], vals[6], vals[7]);
  }
}

// ---------------------------------------------------------------------------
extern "C" void kernel_launch(void* const* d_in, const int* in_sizes, int n_in_args,
                              void* d_out, int out_size, void* d_ws, size_t ws_size,
                              hipStream_t stream) {
  const float* x        = (const float*)d_in[0];
  const int*   nnodes   = (const int*)  d_in[1];
  const float* agg_w    = (const float*)d_in[2];
  const float* agg_b    = (const float*)d_in[3];
  const float* mh_w     = (const float*)d_in[4];
  const float* mh_b     = (const float*)d_in[5];
  const float* shrink_w = (const float*)d_in[6];
  const float* shrink_b = (const float*)d_in[7];
  float* out = (float*)d_out;
  float* ws  = (float*)d_ws;   // 156 floats used

  prep_w2<<<1, 256, 0, stream>>>(shrink_w, shrink_b, mh_w, mh_b, ws);
  copy_head<<<15000, 256, 0, stream>>>((const float4*)x, (float4*)out);
  gal_main<<<10000, 256, 0, stream>>>(x, nnodes, agg_w, agg_b, ws, out);
}